// DBN_57174604644642
// MI455X (gfx1250) — compile-verified
//
#include <hip/hip_runtime.h>
#include <hip/hip_bf16.h>
#include <cstdint>

typedef __attribute__((ext_vector_type(2))) float v2f;
typedef __attribute__((ext_vector_type(4))) float v4f;
typedef __attribute__((ext_vector_type(8))) float v8f;

#define G 64
#define NS_ITERS 12

// Workspace layout (floats):
//  [0,4096)     S      raw second-moment accumulator (64x64)
//  [4096,4160)  colsum per-group sums
//  [4160,8256)  wm     whitening matrix sigma^{-1/2}
//  [8256,8320)  mean
//  [8320,8384)  wmu = wm @ mean
static constexpr int WS_S      = 0;
static constexpr int WS_SUM    = 4096;
static constexpr int WS_WM     = 4160;
static constexpr int WS_MEAN   = 8256;
static constexpr int WS_WMU    = 8320;
static constexpr int WS_FLOATS = 8384;

__global__ void dbn_zero_ws(float* __restrict__ ws) {
    for (int i = threadIdx.x; i < WS_FLOATS; i += blockDim.x) ws[i] = 0.0f;
}

// ---------------------------------------------------------------------------
// Pass 1: S = sum_j x_j x_j^T  and colsum = sum_j x_j, using f32 WMMA.
// Column j of the 64 x m matrix is 64 contiguous floats at in + j*64.
// A-tile (16x4 f32): lanes 0-15 hold K=0,1 ; lanes 16-31 hold K=2,3.
// For X*X^T the B-tile (4x16) register image is identical to the A-tile of
// the same 16-group block, so 4 loaded tiles feed all 16 output tiles.
// ---------------------------------------------------------------------------
__global__ __launch_bounds__(256) void dbn_covar(const float* __restrict__ in,
                                                 float* __restrict__ ws,
                                                 int m) {
    __shared__ float Ssh[G * G];
    __shared__ float Msh[G];
    for (int i = threadIdx.x; i < G * G; i += blockDim.x) Ssh[i] = 0.0f;
    if (threadIdx.x < G) Msh[threadIdx.x] = 0.0f;
    __syncthreads();

    const int lane  = threadIdx.x & 31;
    const int wid   = threadIdx.x >> 5;
    const int nwb   = blockDim.x >> 5;
    const int gwave = blockIdx.x * nwb + wid;
    const int nwv   = gridDim.x * nwb;
    const int half  = lane >> 4;   // 0: K=0,1   1: K=2,3
    const int l     = lane & 15;

    v8f acc[4][4] = {};
    float s[4] = {0.0f, 0.0f, 0.0f, 0.0f};

    const int nchunks = m >> 2;                     // 4 columns per chunk
    for (int ch = gwave; ch < nchunks; ch += nwv) {
        const float* colbase = in + (size_t)(ch * 4 + 2 * half) * G;
        v2f t[4];
#pragma unroll
        for (int r = 0; r < 4; ++r) {
            t[r][0] = colbase[16 * r + l];          // column j0+2h
            t[r][1] = colbase[G + 16 * r + l];      // column j0+2h+1
            s[r] += t[r][0] + t[r][1];
        }
#pragma unroll
        for (int r = 0; r < 4; ++r)
#pragma unroll
            for (int c = 0; c < 4; ++c)
                acc[r][c] = __builtin_amdgcn_wmma_f32_16x16x4_f32(
                    false, t[r], false, t[c], (short)0, acc[r][c], false, false);
    }

    // Reduce wave-private 64x64 accumulators into LDS, then to global.
#pragma unroll
    for (int r = 0; r < 4; ++r) {
        atomicAdd(&Msh[16 * r + l], s[r]);
#pragma unroll
        for (int c = 0; c < 4; ++c)
#pragma unroll
            for (int v = 0; v < 8; ++v) {
                const int M = 16 * r + v + half * 8;   // D layout: vgpr v -> row v / 8+v
                const int N = 16 * c + l;
                atomicAdd(&Ssh[M * G + N], acc[r][c][v]);
            }
    }
    __syncthreads();
    for (int i = threadIdx.x; i < G * G; i += blockDim.x)
        atomicAdd(&ws[WS_S + i], Ssh[i]);
    if (threadIdx.x < G) atomicAdd(&ws[WS_SUM + threadIdx.x], Msh[threadIdx.x]);
}

// ---------------------------------------------------------------------------
// Pass 2 (single block): sigma = S/m - mean*mean^T, then coupled
// Newton-Schulz iteration on A = sigma/tr(sigma):
//   Y_{k+1} = 0.5*(3Y - Y(ZY)),  Z_{k+1} = 0.5*(3Z - (ZY)Z),  Z -> A^{-1/2}
// wm = Z * rsqrt(tr);  wmu = wm @ mean (folded mean correction for pass 3).
// ---------------------------------------------------------------------------
__global__ __launch_bounds__(256) void dbn_newton(float* __restrict__ ws, float inv_m) {
    __shared__ float As[G * G];
    __shared__ float Ys[G * G];
    __shared__ float Zs[G * G];
    __shared__ float Ps[G * G];
    __shared__ float meansh[G];
    __shared__ float trsh;

    const int t = threadIdx.x;
    if (t < G) meansh[t] = ws[WS_SUM + t] * inv_m;
    __syncthreads();
    for (int e = t; e < G * G; e += 256) {
        const int i = e >> 6, j = e & 63;
        As[e] = ws[WS_S + e] * inv_m - meansh[i] * meansh[j];
    }
    __syncthreads();
    if (t == 0) {
        float tr = 0.0f;
        for (int i = 0; i < G; ++i) tr += As[i * G + i];
        trsh = tr;
    }
    __syncthreads();
    const float invtr = 1.0f / trsh;
    for (int e = t; e < G * G; e += 256) {
        const int i = e >> 6, j = e & 63;
        Ys[e] = As[e] * invtr;
        Zs[e] = (i == j) ? 1.0f : 0.0f;
    }
    __syncthreads();

    for (int it = 0; it < NS_ITERS; ++it) {
        for (int e = t; e < G * G; e += 256) {        // P = Z @ Y
            const int i = e >> 6, j = e & 63;
            float p = 0.0f;
            for (int k = 0; k < G; ++k) p += Zs[i * G + k] * Ys[k * G + j];
            Ps[e] = p;
        }
        __syncthreads();
        float yn[16], zn[16];
        int q = 0;
        for (int e = t; e < G * G; e += 256, ++q) {
            const int i = e >> 6, j = e & 63;
            float ay = 0.0f, az = 0.0f;
            for (int k = 0; k < G; ++k) {
                ay += Ys[i * G + k] * Ps[k * G + j];  // Y @ P
                az += Ps[i * G + k] * Zs[k * G + j];  // P @ Z
            }
            yn[q] = 0.5f * (3.0f * Ys[e] - ay);
            zn[q] = 0.5f * (3.0f * Zs[e] - az);
        }
        __syncthreads();
        q = 0;
        for (int e = t; e < G * G; e += 256, ++q) { Ys[e] = yn[q]; Zs[e] = zn[q]; }
        __syncthreads();
    }

    const float sc = rsqrtf(trsh);
    for (int e = t; e < G * G; e += 256) ws[WS_WM + e] = Zs[e] * sc;
    if (t < G) {
        float acc = 0.0f;
        for (int k = 0; k < G; ++k) acc += Zs[t * G + k] * meansh[k];
        ws[WS_WMU + t] = acc * sc;
        ws[WS_MEAN + t] = meansh[t];
    }
}

// ---------------------------------------------------------------------------
// Pass 3: out column j = wm @ x_j - wmu  (mean folded through wm).
// wm held entirely in per-wave VGPRs as pre-swizzled A-tiles; inner loop is
// 1 global_load_b64 (B) + 4 WMMA per K-step. Output stored with 128-bit
// NON-TEMPORAL stores so the L2 (192MB) keeps the input resident.
// ---------------------------------------------------------------------------
__global__ __launch_bounds__(256) void dbn_apply(const float* __restrict__ in,
                                                 const float* __restrict__ ws,
                                                 float* __restrict__ out,
                                                 int m) {
    const int lane  = threadIdx.x & 31;
    const int wid   = threadIdx.x >> 5;
    const int nwb   = blockDim.x >> 5;
    const int gwave = blockIdx.x * nwb + wid;
    const int nwv   = gridDim.x * nwb;
    const int half  = lane >> 4;
    const int l     = lane & 15;

    // Pre-swizzled A-tiles of wm: w[r][s] holds (M = 16r+l, K = 4s+2*half+{0,1}).
    v2f w[4][16];
#pragma unroll
    for (int r = 0; r < 4; ++r) {
        const v2f* wrow = (const v2f*)(ws + WS_WM + (16 * r + l) * G);
#pragma unroll
        for (int s = 0; s < 16; ++s) w[r][s] = wrow[2 * s + half];
    }
    // wmu slices matching the D layout: wmu_r[r][v] = wmu[16r + 8*half + v].
    v8f wmu_r[4];
#pragma unroll
    for (int r = 0; r < 4; ++r)
#pragma unroll
        for (int v = 0; v < 8; ++v)
            wmu_r[r][v] = ws[WS_WMU + 16 * r + 8 * half + v];

    const int ntiles = m >> 4;                       // 16 columns per tile
    for (int tile = gwave; tile < ntiles; tile += nwv) {
        const size_t col = (size_t)tile * 16 + l;    // B/D: N = lane&15
        const v2f* colp = (const v2f*)(in + col * G);
        v8f acc[4] = {};
#pragma unroll
        for (int s = 0; s < 16; ++s) {               // K = 4s + 2*half + comp
            const v2f b = colp[2 * s + half];
#pragma unroll
            for (int r = 0; r < 4; ++r)
                acc[r] = __builtin_amdgcn_wmma_f32_16x16x4_f32(
                    false, w[r][s], false, b, (short)0, acc[r], false, false);
        }
        float* op = out + col * G;
#pragma unroll
        for (int r = 0; r < 4; ++r) {
            float* p = op + 16 * r + 8 * half;
            v4f lo = {acc[r][0] - wmu_r[r][0], acc[r][1] - wmu_r[r][1],
                      acc[r][2] - wmu_r[r][2], acc[r][3] - wmu_r[r][3]};
            v4f hi = {acc[r][4] - wmu_r[r][4], acc[r][5] - wmu_r[r][5],
                      acc[r][6] - wmu_r[r][6], acc[r][7] - wmu_r[r][7]};
            __builtin_nontemporal_store(lo, (v4f*)p);
            __builtin_nontemporal_store(hi, (v4f*)(p + 4));
        }
    }
}

extern "C" void kernel_launch(void* const* d_in, const int* in_sizes, int n_in,
                              void* d_out, int out_size, void* d_ws, size_t ws_size,
                              hipStream_t stream) {
    (void)n_in; (void)out_size; (void)ws_size;
    const float* in = (const float*)d_in[0];
    float* out = (float*)d_out;
    float* ws  = (float*)d_ws;
    const int total = in_sizes[0];       // 8192*4096 = 33554432
    const int m = total / G;             // 524288 columns

    dbn_zero_ws<<<1, 256, 0, stream>>>(ws);
    dbn_covar <<<512, 256, 0, stream>>>(in, ws, m);
    dbn_newton<<<1, 256, 0, stream>>>(ws, 1.0f / (float)m);
    dbn_apply <<<1024, 256, 0, stream>>>(in, ws, out, m);
}